// StockNet_17832704213428
// MI455X (gfx1250) — compile-verified
//
#include <hip/hip_runtime.h>
#include <hip/hip_bf16.h>

typedef float v2f __attribute__((ext_vector_type(2)));
typedef float v8f __attribute__((ext_vector_type(8)));

#define NS 2048
#define NND 12
#define NT 8
#define NE 32768
#define NH 64
#define NEMB 512
#define NGATE 192   // 3*H
#define CHUNK_ROWS 3072              // GRU rows per chunk (24576/8)
#define CHUNK_M (CHUNK_ROWS * 8)     // GEMM rows per chunk
#define BSP 193                      // LDS B-tile stride (float2 units), padded

// ---------------------------------------------------------------------------
// Kernel 1: GI = X[CHUNK_M,512] @ Wih^T  (Wih is [192,512] row-major)
// fp32 WMMA 16x16x4. Each wave: 16 rows x 192 cols (12 n-tiles).
// B chunk staged in LDS as K-pairs (float2) so each fragment is one
// ds_load_b64 straight into the WMMA operand register pair.
// ---------------------------------------------------------------------------
__global__ __launch_bounds__(256) void k_gemm_tweet(
    const float* __restrict__ X, const float* __restrict__ Wih,
    float* __restrict__ GI) {
  __shared__ v2f Bs2[32 * BSP];     // [kpair][n], 49.4 KB
  float* Bf = (float*)Bs2;
  const int tid = threadIdx.x;
  const int wave = tid >> 5, lane = tid & 31;
  const int m = lane & 15, kh = lane >> 4;   // kh selects K-pair {0,1} vs {2,3}
  const int row0 = blockIdx.x * 128 + wave * 16;

  v8f acc[12];
#pragma unroll
  for (int nt = 0; nt < 12; ++nt) acc[nt] = (v8f){0,0,0,0,0,0,0,0};

  for (int k0 = 0; k0 < 512; k0 += 64) {
    __syncthreads();
    for (int idx = tid; idx < 64 * 192; idx += 256) {
      int n = idx >> 6, kk = idx & 63;                       // coalesced over kk
      Bf[((kk >> 1) * BSP + n) * 2 + (kk & 1)] = Wih[n * 512 + k0 + kk];
    }
    __syncthreads();
    const float* arow = X + (size_t)(row0 + m) * 512 + k0 + 2 * kh;
    if (k0 < 448) __builtin_prefetch(arow + 64, 0, 1);       // next A chunk
#pragma unroll 4
    for (int kk = 0; kk < 64; kk += 4) {
      v2f a = *(const v2f*)(arow + kk);                      // A[m][kk+2kh +0/+1]
      const v2f* brow = &Bs2[((kk + 2 * kh) >> 1) * BSP + m];
#pragma unroll
      for (int nt = 0; nt < 12; ++nt) {
        v2f b = brow[nt * 16];                               // one ds_load_b64
        acc[nt] = __builtin_amdgcn_wmma_f32_16x16x4_f32(
            false, a, false, b, (short)0, acc[nt], false, false);
      }
    }
  }
  // D layout: vgpr v, lanes0-15 -> M=v, lanes16-31 -> M=v+8; N = nt*16+m
#pragma unroll
  for (int nt = 0; nt < 12; ++nt) {
#pragma unroll
    for (int v = 0; v < 8; ++v) {
      int rrow = row0 + v + 8 * kh;
      GI[(size_t)rrow * NGATE + nt * 16 + m] = acc[nt][v];
    }
  }
}

// ---------------------------------------------------------------------------
// Kernel 2: tweet GRU recurrence over T=8, sum_t h_t, mask -> Y[row][64]
// 64 threads per GRU row, 4 rows per 256-thread block. whh in LDS (pad 65).
// ---------------------------------------------------------------------------
__global__ __launch_bounds__(256) void k_gru_tweet(
    const float* __restrict__ GI, const float* __restrict__ whh,
    const float* __restrict__ bih, const float* __restrict__ bhh,
    const int* __restrict__ tmask, float* __restrict__ Y, int row_base) {
  __shared__ float sW[192 * 65];
  __shared__ float sH[4][68];
  const int tid = threadIdx.x;
  for (int idx = tid; idx < 192 * 64; idx += 256)
    sW[(idx >> 6) * 65 + (idx & 63)] = whh[idx];
  const int u = tid >> 6, j = tid & 63;
  const int lrow = blockIdx.x * 4 + u;          // chunk-local GRU row
  const int grow = row_base + lrow;             // global row = s*12+nd
  const float* gi = GI + (size_t)lrow * 8 * NGATE;
  sH[u][j] = 0.f;
  const float b0 = bih[j], b1 = bih[64 + j], b2 = bih[128 + j];
  const float c0 = bhh[j], c1 = bhh[64 + j], c2 = bhh[128 + j];
  const float* w0 = &sW[j * 65];
  const float* w1 = &sW[(64 + j) * 65];
  const float* w2 = &sW[(128 + j) * 65];
  float sum = 0.f;
  __syncthreads();
  for (int t = 0; t < 8; ++t) {
    float gi0 = gi[t * NGATE + j] + b0;
    float gi1 = gi[t * NGATE + 64 + j] + b1;
    float gi2 = gi[t * NGATE + 128 + j] + b2;
    float gh0 = c0, gh1 = c1, gh2 = c2;
    float hold = sH[u][j];
#pragma unroll 8
    for (int i = 0; i < 64; ++i) {
      float hi = sH[u][i];
      gh0 += hi * w0[i]; gh1 += hi * w1[i]; gh2 += hi * w2[i];
    }
    float r = 1.f / (1.f + expf(-(gi0 + gh0)));
    float z = 1.f / (1.f + expf(-(gi1 + gh1)));
    float n = tanhf(gi2 + r * gh2);
    float hn = (1.f - z) * n + z * hold;
    __syncthreads();
    sH[u][j] = hn; sum += hn;
    __syncthreads();
  }
  float msk = (tmask[grow] > 0) ? 1.f : 0.f;
  Y[(size_t)grow * NH + j] = sum * msk;
}

// ---------------------------------------------------------------------------
// Kernel 3: price GRU over price[:, w:w+8, :3], sum_t h_t -> XP[s][64]
// ---------------------------------------------------------------------------
__global__ __launch_bounds__(256) void k_gru_price(
    const float* __restrict__ price, const float* __restrict__ wih,
    const float* __restrict__ whh, const float* __restrict__ bih,
    const float* __restrict__ bhh, float* __restrict__ XP, int w) {
  __shared__ float sW[192 * 65];
  __shared__ float sH[4][68];
  const int tid = threadIdx.x;
  for (int idx = tid; idx < 192 * 64; idx += 256)
    sW[(idx >> 6) * 65 + (idx & 63)] = whh[idx];
  const int u = tid >> 6, j = tid & 63;
  const int s = blockIdx.x * 4 + u;
  const float a00 = wih[j * 3], a01 = wih[j * 3 + 1], a02 = wih[j * 3 + 2];
  const float a10 = wih[(64 + j) * 3], a11 = wih[(64 + j) * 3 + 1], a12 = wih[(64 + j) * 3 + 2];
  const float a20 = wih[(128 + j) * 3], a21 = wih[(128 + j) * 3 + 1], a22 = wih[(128 + j) * 3 + 2];
  const float b0 = bih[j], b1 = bih[64 + j], b2 = bih[128 + j];
  const float c0 = bhh[j], c1 = bhh[64 + j], c2 = bhh[128 + j];
  const float* w0 = &sW[j * 65];
  const float* w1 = &sW[(64 + j) * 65];
  const float* w2 = &sW[(128 + j) * 65];
  sH[u][j] = 0.f;
  float sum = 0.f;
  __syncthreads();
  for (int t = 0; t < 8; ++t) {
    const float* pr = price + (size_t)s * 15 * 3 + (w + t) * 3;
    float x0 = pr[0], x1 = pr[1], x2 = pr[2];
    float gi0 = b0 + a00 * x0 + a01 * x1 + a02 * x2;
    float gi1 = b1 + a10 * x0 + a11 * x1 + a12 * x2;
    float gi2 = b2 + a20 * x0 + a21 * x1 + a22 * x2;
    float gh0 = c0, gh1 = c1, gh2 = c2;
    float hold = sH[u][j];
#pragma unroll 8
    for (int i = 0; i < 64; ++i) {
      float hi = sH[u][i];
      gh0 += hi * w0[i]; gh1 += hi * w1[i]; gh2 += hi * w2[i];
    }
    float r = 1.f / (1.f + expf(-(gi0 + gh0)));
    float z = 1.f / (1.f + expf(-(gi1 + gh1)));
    float n = tanhf(gi2 + r * gh2);
    float hn = (1.f - z) * n + z * hold;
    __syncthreads();
    sH[u][j] = hn; sum += hn;
    __syncthreads();
  }
  XP[(size_t)s * NH + j] = sum;
}

// ---------------------------------------------------------------------------
// Kernel 4: text-window GRU over y[:, w:w+5, :], sum_t h_t -> TXT[s][64]
// ---------------------------------------------------------------------------
__global__ __launch_bounds__(256) void k_gru_txt(
    const float* __restrict__ Y, const float* __restrict__ wih,
    const float* __restrict__ whh, const float* __restrict__ bih,
    const float* __restrict__ bhh, float* __restrict__ TXT, int w) {
  __shared__ float sW[192 * 65];
  __shared__ float sH[4][68];
  __shared__ float sX[4][68];
  const int tid = threadIdx.x;
  for (int idx = tid; idx < 192 * 64; idx += 256)
    sW[(idx >> 6) * 65 + (idx & 63)] = whh[idx];
  const int u = tid >> 6, j = tid & 63;
  const int s = blockIdx.x * 4 + u;
  const float b0 = bih[j], b1 = bih[64 + j], b2 = bih[128 + j];
  const float c0 = bhh[j], c1 = bhh[64 + j], c2 = bhh[128 + j];
  const float* w0 = &sW[j * 65];
  const float* w1 = &sW[(64 + j) * 65];
  const float* w2 = &sW[(128 + j) * 65];
  const float* i0 = wih + j * 64;
  const float* i1 = wih + (64 + j) * 64;
  const float* i2 = wih + (128 + j) * 64;
  sH[u][j] = 0.f;
  float sum = 0.f;
  __syncthreads();
  for (int t = 0; t < 5; ++t) {
    sX[u][j] = Y[(size_t)(s * NND + w + t) * NH + j];
    __syncthreads();
    float gi0 = b0, gi1 = b1, gi2 = b2;
    float gh0 = c0, gh1 = c1, gh2 = c2;
    float hold = sH[u][j];
#pragma unroll 8
    for (int i = 0; i < 64; ++i) {
      float xi = sX[u][i];
      gi0 += xi * i0[i]; gi1 += xi * i1[i]; gi2 += xi * i2[i];
      float hi = sH[u][i];
      gh0 += hi * w0[i]; gh1 += hi * w1[i]; gh2 += hi * w2[i];
    }
    float r = 1.f / (1.f + expf(-(gi0 + gh0)));
    float z = 1.f / (1.f + expf(-(gi1 + gh1)));
    float n = tanhf(gi2 + r * gh2);
    float hn = (1.f - z) * n + z * hold;
    __syncthreads();
    sH[u][j] = hn; sum += hn;
    __syncthreads();
  }
  TXT[(size_t)s * NH + j] = sum;
}

// ---------------------------------------------------------------------------
// Kernel 5: bilinear  comb[s,k] = tanh( sum_ij txt_i W[k,i,j] x_j + b_k )
// ---------------------------------------------------------------------------
__global__ __launch_bounds__(256) void k_bilinear(
    const float* __restrict__ TXT, const float* __restrict__ XP,
    const float* __restrict__ bw, const float* __restrict__ bb,
    float* __restrict__ COMB) {
  __shared__ float sT[4][68], sX[4][68];
  const int tid = threadIdx.x;
  const int u = tid >> 6, j = tid & 63;
  const int s = blockIdx.x * 4 + u;
  sT[u][j] = TXT[(size_t)s * NH + j];
  sX[u][j] = XP[(size_t)s * NH + j];
  __syncthreads();
  float acc = bb[j];
  for (int i = 0; i < 64; ++i) {
    const float* wrow = bw + ((size_t)j * 64 + i) * 64;
    float part = 0.f;
#pragma unroll 8
    for (int jj = 0; jj < 64; ++jj) part += wrow[jj] * sX[u][jj];
    acc += sT[u][i] * part;
  }
  COMB[(size_t)s * NH + j] = tanhf(acc);
}

// ---------------------------------------------------------------------------
// Kernel 6A: GPRE = comb @ wself + b ; HREL[r] = comb @ wrel[r]
// ---------------------------------------------------------------------------
__global__ __launch_bounds__(256) void k_gnnA(
    const float* __restrict__ COMB, const float* __restrict__ wself,
    const float* __restrict__ wrel, const float* __restrict__ gb,
    float* __restrict__ GPRE, float* __restrict__ HREL) {
  __shared__ float sC[4][68];
  const int tid = threadIdx.x;
  const int u = tid >> 6, j = tid & 63;
  const int s = blockIdx.x * 4 + u;
  sC[u][j] = COMB[(size_t)s * NH + j];
  __syncthreads();
  float acc = gb[j], h0 = 0.f, h1 = 0.f, h2 = 0.f;
#pragma unroll 4
  for (int i = 0; i < 64; ++i) {
    float ci = sC[u][i];
    acc += ci * wself[i * 64 + j];
    h0 += ci * wrel[(0 * 64 + i) * 64 + j];
    h1 += ci * wrel[(1 * 64 + i) * 64 + j];
    h2 += ci * wrel[(2 * 64 + i) * 64 + j];
  }
  GPRE[(size_t)s * NH + j] = acc;
  HREL[((size_t)0 * NS + s) * NH + j] = h0;
  HREL[((size_t)1 * NS + s) * NH + j] = h1;
  HREL[((size_t)2 * NS + s) * NH + j] = h2;
}

// ---------------------------------------------------------------------------
// CSR build (deterministic segment_sum): count / serial scan / stable fill
// ---------------------------------------------------------------------------
__global__ void k_count(const int* __restrict__ dst, int* __restrict__ counts) {
  int v = blockIdx.x * 256 + threadIdx.x;
  int c = 0;
  for (int e = 0; e < NE; ++e) c += (dst[e] == v) ? 1 : 0;
  counts[v] = c;
}
__global__ void k_scan(const int* __restrict__ counts, int* __restrict__ rowptr) {
  if (blockIdx.x == 0 && threadIdx.x == 0) {
    int acc = 0;
    for (int v = 0; v < NS; ++v) { rowptr[v] = acc; acc += counts[v]; }
    rowptr[NS] = acc;
  }
}
__global__ void k_fill(const int* __restrict__ dst, const int* __restrict__ rowptr,
                       int* __restrict__ elist) {
  int v = blockIdx.x * 256 + threadIdx.x;
  int p = rowptr[v];
  for (int e = 0; e < NE; ++e)
    if (dst[e] == v) elist[p++] = e;   // ascending e: deterministic sum order
}

// ---------------------------------------------------------------------------
// Kernel 6B: g = tanh(GPRE[v] + sum_{e: dst==v} HREL[type[e]][src[e]])
// ---------------------------------------------------------------------------
__global__ __launch_bounds__(256) void k_gather(
    const float* __restrict__ GPRE, const float* __restrict__ HREL,
    const int* __restrict__ rowptr, const int* __restrict__ elist,
    const int* __restrict__ esrc, const int* __restrict__ etype,
    float* __restrict__ GBUF) {
  const int tid = threadIdx.x;
  const int u = tid >> 6, j = tid & 63;
  const int v = blockIdx.x * 4 + u;
  float acc = GPRE[(size_t)v * NH + j];
  const int beg = rowptr[v], end = rowptr[v + 1];
  for (int p = beg; p < end; ++p) {
    int e = elist[p];
    int sv = esrc[e];
    int r = etype[e];
    acc += HREL[((size_t)r * NS + sv) * NH + j];
  }
  GBUF[(size_t)v * NH + j] = tanhf(acc);
}

// ---------------------------------------------------------------------------
// Kernel 7: head  fc1 -> LN -> relu -> fc2 -> LN -> relu -> mlp1 relu -> mlp2
// ---------------------------------------------------------------------------
__global__ __launch_bounds__(256) void k_head(
    const float* __restrict__ COMB, const float* __restrict__ GBUF,
    const float* __restrict__ fc1w, const float* __restrict__ fc1b,
    const float* __restrict__ ln1g, const float* __restrict__ ln1b,
    const float* __restrict__ fc2w, const float* __restrict__ fc2b,
    const float* __restrict__ ln2g, const float* __restrict__ ln2b,
    const float* __restrict__ mw1, const float* __restrict__ mb1,
    const float* __restrict__ mw2, const float* __restrict__ mb2,
    float* __restrict__ out, int w) {
  __shared__ float sC[4][68], sG[4][68], sT[4][68];
  const int tid = threadIdx.x;
  const int u = tid >> 6, j = tid & 63;
  const int s = blockIdx.x * 4 + u;
  sC[u][j] = COMB[(size_t)s * NH + j];
  sG[u][j] = GBUF[(size_t)s * NH + j];
  __syncthreads();
  // fc1: [comb, g] @ fc1_w.T + b    (fc1_w [64,128])
  float a = fc1b[j];
  {
    const float* wr = fc1w + j * 128;
#pragma unroll 8
    for (int i = 0; i < 64; ++i) a += sC[u][i] * wr[i] + sG[u][i] * wr[64 + i];
  }
  // LN1 + relu
  sT[u][j] = a; __syncthreads();
  float m = 0.f, m2 = 0.f;
  for (int i = 0; i < 64; ++i) { float v_ = sT[u][i]; m += v_; m2 += v_ * v_; }
  m *= (1.f / 64.f); m2 = m2 * (1.f / 64.f) - m * m;
  float x = (a - m) * rsqrtf(m2 + 1e-5f) * ln1g[j] + ln1b[j];
  x = fmaxf(x, 0.f);
  __syncthreads();
  sT[u][j] = x; __syncthreads();
  // fc2
  float a2 = fc2b[j];
  {
    const float* wr = fc2w + j * 64;
#pragma unroll 8
    for (int i = 0; i < 64; ++i) a2 += sT[u][i] * wr[i];
  }
  __syncthreads();
  sT[u][j] = a2; __syncthreads();
  m = 0.f; m2 = 0.f;
  for (int i = 0; i < 64; ++i) { float v_ = sT[u][i]; m += v_; m2 += v_ * v_; }
  m *= (1.f / 64.f); m2 = m2 * (1.f / 64.f) - m * m;
  x = (a2 - m) * rsqrtf(m2 + 1e-5f) * ln2g[j] + ln2b[j];
  x = fmaxf(x, 0.f);
  __syncthreads();
  sT[u][j] = x; __syncthreads();
  // mlp1 relu
  float h1 = mb1[j];
  {
    const float* wr = mw1 + j * 64;
#pragma unroll 8
    for (int i = 0; i < 64; ++i) h1 += sT[u][i] * wr[i];
  }
  h1 = fmaxf(h1, 0.f);
  __syncthreads();
  sT[u][j] = h1 * mw2[j]; __syncthreads();
  if (j == 0) {
    float acc = mb2[0];
    for (int i = 0; i < 64; ++i) acc += sT[u][i];
    out[w * NS + s] = acc;
  }
}

// ---------------------------------------------------------------------------
extern "C" void kernel_launch(void* const* d_in, const int* in_sizes, int n_in,
                              void* d_out, int out_size, void* d_ws, size_t ws_size,
                              hipStream_t stream) {
  const float* text  = (const float*)d_in[0];
  const float* price = (const float*)d_in[1];
  const int*   tmask = (const int*)d_in[2];
  const int*   eidx  = (const int*)d_in[3];   // [2, E]: row0 src, row1 dst
  const int*   etype = (const int*)d_in[4];

  int pi = 5;
  auto F = [&](int k) { return (const float*)d_in[k]; };
  // grup (price GRU)
  const float* gp_wih = F(pi++); const float* gp_whh = F(pi++);
  const float* gp_bih = F(pi++); const float* gp_bhh = F(pi++);
  // grtw (tweet GRU)
  const float* tw_wih = F(pi++); const float* tw_whh = F(pi++);
  const float* tw_bih = F(pi++); const float* tw_bhh = F(pi++);
  // grut (text-window GRU)
  const float* gt_wih = F(pi++); const float* gt_whh = F(pi++);
  const float* gt_bih = F(pi++); const float* gt_bhh = F(pi++);
  pi += 18;   // attnp/attnw/attnt: softmax over singleton axis -> unused
  const float* bil_w = F(pi++); const float* bil_b = F(pi++);
  const float* wself = F(pi++); const float* wrel  = F(pi++);
  const float* gnnb  = F(pi++);
  const float* fc1w  = F(pi++); const float* fc1b  = F(pi++);
  const float* ln1g  = F(pi++); const float* ln1b  = F(pi++);
  const float* fc2w  = F(pi++); const float* fc2b  = F(pi++);
  const float* ln2g  = F(pi++); const float* ln2b  = F(pi++);
  const float* mw1   = F(pi++); const float* mb1   = F(pi++);
  const float* mw2   = F(pi++); const float* mb2   = F(pi++);

  // ---- workspace layout (floats) ----
  float* ws = (float*)d_ws;
  size_t off = 0;
  auto alloc = [&](size_t n) { float* p = ws + off; off += (n + 63) & ~(size_t)63; return p; };
  float* GI   = alloc((size_t)CHUNK_M * NGATE);     // 4.72M floats
  float* Y    = alloc((size_t)NS * NND * NH);       // 1.57M floats
  float* XP   = alloc((size_t)NS * NH);
  float* TXT  = alloc((size_t)NS * NH);
  float* COMB = alloc((size_t)NS * NH);
  float* GPRE = alloc((size_t)NS * NH);
  float* GBUF = alloc((size_t)NS * NH);
  float* HREL = alloc((size_t)3 * NS * NH);
  int* counts = (int*)alloc(NS);
  int* rowptr = (int*)alloc(NS + 64);
  int* elist  = (int*)alloc(NE);

  const int* esrc = eidx;
  const int* edst = eidx + NE;

  // CSR for deterministic segment_sum (edge graph is window-invariant)
  k_count<<<NS / 256, 256, 0, stream>>>(edst, counts);
  k_scan<<<1, 32, 0, stream>>>(counts, rowptr);
  k_fill<<<NS / 256, 256, 0, stream>>>(edst, rowptr, elist);

  // Tweet encoder: WMMA GEMM + GRU recurrence, chunked over M
  for (int c = 0; c < 8; ++c) {
    k_gemm_tweet<<<CHUNK_M / 128, 256, 0, stream>>>(
        text + (size_t)c * CHUNK_M * NEMB, tw_wih, GI);
    k_gru_tweet<<<CHUNK_ROWS / 4, 256, 0, stream>>>(
        GI, tw_whh, tw_bih, tw_bhh, tmask, Y, c * CHUNK_ROWS);
  }

  // Per-window pipeline
  for (int w = 0; w < 8; ++w) {
    k_gru_price<<<NS / 4, 256, 0, stream>>>(price, gp_wih, gp_whh, gp_bih, gp_bhh, XP, w);
    k_gru_txt<<<NS / 4, 256, 0, stream>>>(Y, gt_wih, gt_whh, gt_bih, gt_bhh, TXT, w);
    k_bilinear<<<NS / 4, 256, 0, stream>>>(TXT, XP, bil_w, bil_b, COMB);
    k_gnnA<<<NS / 4, 256, 0, stream>>>(COMB, wself, wrel, gnnb, GPRE, HREL);
    k_gather<<<NS / 4, 256, 0, stream>>>(GPRE, HREL, rowptr, elist, esrc, etype, GBUF);
    k_head<<<NS / 4, 256, 0, stream>>>(COMB, GBUF, fc1w, fc1b, ln1g, ln1b,
                                       fc2w, fc2b, ln2g, ln2b, mw1, mb1, mw2, mb2,
                                       (float*)d_out, w);
  }
}